// GaussianInfoNCELoss_9637906612629
// MI455X (gfx1250) — compile-verified
//
#include <hip/hip_runtime.h>
#include <hip/hip_bf16.h>

// GaussianInfoNCE loss, fused GEMM + online logsumexp for MI455X (gfx1250).
//
// loss = mean_i [ logsumexp_j z_ij - z_ii ],  z_ij = x_i . y_j - 0.5*|y_j|^2
// (the -0.5|x_i|^2 term cancels between lse and the diagonal).
//
// K = 8192, D = 128. GEMM: 2*K*K*D = 17.2 GFLOP, inputs 8 MB (L2-resident).
// fp32 matrix path: V_WMMA_F32_16X16X4_F32, 32 k-steps per 16x16 tile.
// Each wave now carries TWO 16-row A panels and fires two WMMAs (independent
// accumulator chains) per B fragment load: 2x FLOP per byte fetched from L2
// and interleaved WMMA chains to hide matrix-pipe latency.

typedef __attribute__((ext_vector_type(2))) float v2f;
typedef __attribute__((ext_vector_type(8))) float v8f;

#define DIMD 128
#define SCALE 0.5f  // GAMMA / TEMPERATURE = 1/(2*sigma^2) / T
#define ROWS_PER_BLOCK 32

// ---------------------------------------------------------------------------
// Kernel 1: per-row precompute.
//   ysq_s[i]  = SCALE * |y_i|^2
//   zdiag[i]  = 2*SCALE * (x_i . y_i) - SCALE * |y_i|^2
// Also zeroes the output accumulator (harness poisons d_out).
// ---------------------------------------------------------------------------
__global__ void __launch_bounds__(256)
precomp_kernel(const float* __restrict__ x, const float* __restrict__ y,
               float* __restrict__ ysq_s, float* __restrict__ zdiag,
               float* __restrict__ out, int K) {
  const int i = blockIdx.x * blockDim.x + threadIdx.x;
  if (i == 0) *out = 0.0f;
  if (i >= K) return;
  const float4* __restrict__ xr = (const float4*)(x + (size_t)i * DIMD);
  const float4* __restrict__ yr = (const float4*)(y + (size_t)i * DIMD);
  float dot = 0.0f, sq = 0.0f;
#pragma unroll
  for (int d = 0; d < DIMD / 4; ++d) {
    const float4 xv = xr[d];
    const float4 yv = yr[d];
    dot += xv.x * yv.x + xv.y * yv.y + xv.z * yv.z + xv.w * yv.w;
    sq  += yv.x * yv.x + yv.y * yv.y + yv.z * yv.z + yv.w * yv.w;
  }
  ysq_s[i] = SCALE * sq;
  zdiag[i] = 2.0f * SCALE * dot - SCALE * sq;
}

__device__ __forceinline__ void lse_combine(float& m, float& s, float mo, float so) {
  const float nm = fmaxf(m, mo);
  s = s * __expf(m - nm) + so * __expf(mo - nm);
  m = nm;
}

// ---------------------------------------------------------------------------
// Kernel 2: one block = one 32-row tile (two 16-row WMMA panels). 8 waves
// sweep the K/16 column tiles with stride 8. Per column tile:
//   32 k-steps x { 1 global_load_b64 (B frag), 2 x V_WMMA_F32_16X16X4_F32 }
// then 16 online-softmax updates per lane.
//
// fp32 16x16x4 A layout: lanes 0-15 hold rows 0-15 with K={0,1} in v0/v1;
// lanes 16-31 hold the same rows with K={2,3}. B (4x16, = y^T tile) mirrors
// this with N=lane&15. C/D layout: lane holds column N=lane&15, rows r
// (lanes<16) / r+8 (lanes>=16) in vgpr r.
// ---------------------------------------------------------------------------
__global__ void __launch_bounds__(256)
lse_kernel(const float* __restrict__ x, const float* __restrict__ y,
           const float* __restrict__ ysq_s, const float* __restrict__ zdiag,
           float* __restrict__ out, int K) {
  const int ntiles = K >> 4;                   // 512 column tiles
  const int row0 = blockIdx.x * ROWS_PER_BLOCK;
  const int wave = threadIdx.x >> 5;           // 8 waves
  const int lane = threadIdx.x & 31;
  const int mrow = lane & 15;                  // row (A) / column (B,C) index
  const int koff = (lane >> 4) << 1;           // K sub-offset {0,2}

  // Two A panels: rows [row0, row0+16) and [row0+16, row0+32). 128 VGPRs.
  const float* __restrict__ xrow0 = x + (size_t)(row0 + mrow) * DIMD + koff;
  const float* __restrict__ xrow1 = xrow0 + (size_t)16 * DIMD;
  v2f a0[DIMD / 4], a1[DIMD / 4];
#pragma unroll
  for (int kk = 0; kk < DIMD / 4; ++kk) {
    a0[kk] = *(const v2f*)(xrow0 + 4 * kk);
    a1[kk] = *(const v2f*)(xrow1 + 4 * kk);
  }

  // Online logsumexp state: [0..7] = panel-0 rows, [8..15] = panel-1 rows.
  float m[16], s[16];
#pragma unroll
  for (int r = 0; r < 16; ++r) { m[r] = -__builtin_inff(); s[r] = 0.0f; }

  for (int t = wave; t < ntiles; t += 8) {
    const int col0 = t << 4;
    const float* __restrict__ yrow = y + (size_t)(col0 + mrow) * DIMD + koff;
    if (t + 8 < ntiles)  // stage this wave's next B tile into cache
      __builtin_prefetch(y + (size_t)(((t + 8) << 4) + mrow) * DIMD + koff, 0, 1);

    v8f c0 = {}, c1 = {};
#pragma unroll
    for (int kk = 0; kk < DIMD / 4; ++kk) {
      const v2f b = *(const v2f*)(yrow + 4 * kk);  // shared by both panels
      c0 = __builtin_amdgcn_wmma_f32_16x16x4_f32(
          false, a0[kk], false, b, (short)0, c0, false, false);
      c1 = __builtin_amdgcn_wmma_f32_16x16x4_f32(
          false, a1[kk], false, b, (short)0, c1, false, false);
    }

    const float bias = ysq_s[col0 + mrow];  // SCALE*|y_col|^2, shared by panels
#pragma unroll
    for (int r = 0; r < 8; ++r) {
      {
        const float v = c0[r] - bias;
        const float nm = fmaxf(m[r], v);
        s[r] = s[r] * __expf(m[r] - nm) + __expf(v - nm);
        m[r] = nm;
      }
      {
        const float v = c1[r] - bias;
        const float nm = fmaxf(m[r + 8], v);
        s[r + 8] = s[r + 8] * __expf(m[r + 8] - nm) + __expf(v - nm);
        m[r + 8] = nm;
      }
    }
  }

  // Butterfly (max,sum) combine across the 16 columns within each lane half.
#pragma unroll
  for (int off = 1; off <= 8; off <<= 1) {
#pragma unroll
    for (int r = 0; r < 16; ++r) {
      const float mo = __shfl_xor(m[r], off, 32);
      const float so = __shfl_xor(s[r], off, 32);
      lse_combine(m[r], s[r], mo, so);
    }
  }

  // Cross-wave combine via LDS: wave w contributes (m,s) for 32 rows.
  __shared__ float lm[8][ROWS_PER_BLOCK];
  __shared__ float ls[8][ROWS_PER_BLOCK];
  __shared__ float lc[ROWS_PER_BLOCK];
  if (mrow == 0) {
    const int rbase = (lane >> 4) << 3;  // lane 0 -> rows 0-7, lane 16 -> 8-15
#pragma unroll
    for (int r = 0; r < 8; ++r) {
      lm[wave][rbase + r] = m[r];            // panel 0
      ls[wave][rbase + r] = s[r];
      lm[wave][16 + rbase + r] = m[r + 8];   // panel 1
      ls[wave][16 + rbase + r] = s[r + 8];
    }
  }
  __syncthreads();

  if (threadIdx.x < ROWS_PER_BLOCK) {
    const int rr = threadIdx.x;
    float M = lm[0][rr], S = ls[0][rr];
#pragma unroll
    for (int w = 1; w < 8; ++w) lse_combine(M, S, lm[w][rr], ls[w][rr]);
    const float g = M + __logf(S);          // logsumexp_j z_{row,j}
    lc[rr] = g - zdiag[row0 + rr];          // per-row loss contribution
  }
  __syncthreads();

  if (threadIdx.x == 0) {
    float acc = 0.0f;
#pragma unroll
    for (int rr = 0; rr < ROWS_PER_BLOCK; ++rr) acc += lc[rr];
    atomicAdd(out, acc / (float)K);
  }
}

// ---------------------------------------------------------------------------
extern "C" void kernel_launch(void* const* d_in, const int* in_sizes, int n_in,
                              void* d_out, int out_size, void* d_ws, size_t ws_size,
                              hipStream_t stream) {
  (void)n_in; (void)out_size; (void)ws_size;
  const float* x = (const float*)d_in[0];   // features_nc, (K, 128) fp32
  const float* y = (const float*)d_in[1];   // features_c,  (K, 128) fp32
  const int K = in_sizes[0] / DIMD;         // 8192

  float* ysq_s = (float*)d_ws;              // K floats
  float* zdiag = ysq_s + K;                 // K floats
  float* out = (float*)d_out;               // scalar loss

  precomp_kernel<<<(K + 255) / 256, 256, 0, stream>>>(x, y, ysq_s, zdiag, out, K);
  lse_kernel<<<K / ROWS_PER_BLOCK, 256, 0, stream>>>(x, y, ysq_s, zdiag, out, K);
}